// IDWT_Fixed_67044439490878
// MI455X (gfx1250) — compile-verified
//
#include <hip/hip_runtime.h>
#include <hip/hip_bf16.h>

typedef float v4f __attribute__((ext_vector_type(4)));
typedef int   v4i __attribute__((ext_vector_type(4)));

// Async global->LDS path is a device-side decision only (host pass lacks the
// amdgcn builtins); launch geometry is identical for both paths.
#if defined(__AMDGCN__) && __has_builtin(__builtin_amdgcn_global_load_async_to_lds_b128) && __has_builtin(__builtin_amdgcn_s_wait_asynccnt)
#define IDWT_ASYNC 1
#else
#define IDWT_ASYNC 0
#endif

#if IDWT_ASYNC
typedef __attribute__((address_space(1))) v4i* global_v4i_ptr;  // prints as "int4 __device__*"
typedef __attribute__((address_space(3))) v4i* lds_v4i_ptr;     // prints as "int4 __shared__*"
#endif

namespace {

constexpr int BLOCK           = 256;   // 8 wave32s
constexpr int TILES_PER_BLOCK = 4;     // tiles processed per block (pipelined)
constexpr int TILE_FLOATS     = 1024;  // 4 input rows x 256 cols per input per tile
// Fixed problem shape from the reference: B=8, C=64, H=256, W=256.

__device__ __forceinline__ void idwt_compute_store(
    v4f ll, v4f lh, v4f hl, v4f hh,
    float* __restrict__ out, int globalTile, int tid)
{
    // x1 = LL - (LH+HL-HH)/2 ; x2 = LL - (LH-HL+HH)/2
    // x3 = LL + (LH-HL-HH)/2 ; x4 = LL + (LH+HL+HH)/2
    v4f p = 0.5f * (lh + hl);
    v4f q = 0.5f * (lh - hl);
    v4f r = 0.5f * hh;
    v4f x1 = ll - p + r;   // even row, even col
    v4f x2 = ll - q - r;   // odd  row, even col
    v4f x3 = ll + q - r;   // even row, odd  col
    v4f x4 = ll + p + r;   // odd  row, odd  col

    int row = globalTile * 4 + (tid >> 6);   // input row in [0, B*C*H)
    int cir = tid & 63;                      // which chunk-of-4 within the input row
    int bc  = row >> 8;                      // H = 256
    int h   = row & 255;
    long evenBase = (long)bc * (512 * 512) + (long)(2 * h) * 512 + (long)cir * 8;
    float* pe = out + evenBase;              // even output row (x1,x3 interleave)
    float* po = pe + 512;                    // odd  output row (x2,x4 interleave)

    v4f e0 = {x1.x, x3.x, x1.y, x3.y};
    v4f e1 = {x1.z, x3.z, x1.w, x3.w};
    v4f o0 = {x2.x, x4.x, x2.y, x4.y};
    v4f o1 = {x2.z, x4.z, x2.w, x4.w};
    // Output is written once, never re-read: stream past L2 with NT stores.
    __builtin_nontemporal_store(e0, (v4f*)pe);
    __builtin_nontemporal_store(e1, (v4f*)(pe + 4));
    __builtin_nontemporal_store(o0, (v4f*)po);
    __builtin_nontemporal_store(o1, (v4f*)(po + 4));
}

__global__ __launch_bounds__(BLOCK) void idwt_kernel(
    const float* __restrict__ LL, const float* __restrict__ LH,
    const float* __restrict__ HL, const float* __restrict__ HH,
    float* __restrict__ out)
{
    const int tid   = threadIdx.x;
    const int tile0 = blockIdx.x * TILES_PER_BLOCK;

#if IDWT_ASYNC
    // Double-buffered async global->LDS pipeline (ASYNCcnt-tracked, CDNA5 path).
    __shared__ float lds[2][4][TILE_FLOATS];   // 32 KB of 320 KB/WGP
    const float* srcs[4] = {LL, LH, HL, HH};

    auto issue = [&](int t, int buf) {
        long off = (long)(tile0 + t) * TILE_FLOATS + (long)tid * 4;
        #pragma unroll
        for (int i = 0; i < 4; ++i) {
            float* gp = const_cast<float*>(srcs[i] + off);
            float* lp = &lds[buf][i][tid * 4];
            __builtin_amdgcn_global_load_async_to_lds_b128(
                (global_v4i_ptr)gp, (lds_v4i_ptr)lp, 0, 0);
        }
    };

    issue(0, 0);
    #pragma unroll
    for (int t = 0; t < TILES_PER_BLOCK; ++t) {
        if (t + 1 < TILES_PER_BLOCK) {
            issue(t + 1, (t + 1) & 1);
            // Async loads complete in order: <=4 outstanding means tile t landed.
            __builtin_amdgcn_s_wait_asynccnt(4);
        } else {
            __builtin_amdgcn_s_wait_asynccnt(0);
        }
        __syncthreads();
        const int b = t & 1;
        v4f ll = *(const v4f*)(&lds[b][0][tid * 4]);
        v4f lh = *(const v4f*)(&lds[b][1][tid * 4]);
        v4f hl = *(const v4f*)(&lds[b][2][tid * 4]);
        v4f hh = *(const v4f*)(&lds[b][3][tid * 4]);
        idwt_compute_store(ll, lh, hl, hh, out, tile0 + t, tid);
        __syncthreads();   // buffer b is reused by the issue() two tiles ahead
    }
#else
    // Direct streaming path: NT b128 loads, same per-thread work assignment.
    #pragma unroll
    for (int t = 0; t < TILES_PER_BLOCK; ++t) {
        long off = (long)(tile0 + t) * TILE_FLOATS + (long)tid * 4;
        v4f ll = __builtin_nontemporal_load((const v4f*)(LL + off));
        v4f lh = __builtin_nontemporal_load((const v4f*)(LH + off));
        v4f hl = __builtin_nontemporal_load((const v4f*)(HL + off));
        v4f hh = __builtin_nontemporal_load((const v4f*)(HH + off));
        idwt_compute_store(ll, lh, hl, hh, out, tile0 + t, tid);
    }
#endif
}

} // namespace

extern "C" void kernel_launch(void* const* d_in, const int* in_sizes, int n_in,
                              void* d_out, int out_size, void* d_ws, size_t ws_size,
                              hipStream_t stream) {
    const float* LL = (const float*)d_in[0];
    const float* LH = (const float*)d_in[1];
    const float* HL = (const float*)d_in[2];
    const float* HH = (const float*)d_in[3];
    float* out = (float*)d_out;

    const int n       = in_sizes[0];               // B*C*H*W = 33,554,432
    const int nTiles  = n / TILE_FLOATS;           // 32768
    const int nBlocks = nTiles / TILES_PER_BLOCK;  // 8192 blocks x 256 threads

    idwt_kernel<<<nBlocks, BLOCK, 0, stream>>>(LL, LH, HL, HH, out);
}